// QuantumKernel_65481071409030
// MI455X (gfx1250) — compile-verified
//
#include <hip/hip_runtime.h>
#include <math.h>

// ---------------------------------------------------------------------------
// Quantum state-overlap kernel for MI455X (gfx1250).
//
// psi = product state (all-real) permuted by the CNOT ring:
//   psi[idx] = psi_pre[Q(idx)],  Q(b) = b ^ (b>>1) ^ ((b&1)*0x300000)  (22 bits)
//   psi_pre[m] = prod_k f_k(bit_k(m)),  f_k(0)=cos(th[21-k]/2), f_k(1)=sin(...)
// out[4b+2i+j] = | y[4b+2i]*x[4b+j] + y[4b+2i+1]*x[4b+2+j] |
// Batched 2x2 matmuls are packed 8-per-V_WMMA_F32_16X16X4_F32 (block diagonal).
// Bandwidth-bound: 16 MiB f32 output -> ~0.7 us floor at 23.3 TB/s.
// ---------------------------------------------------------------------------

#define NW 22
#define NGROUP (1 << (NW - 2))   // 2^20 groups of 4 outputs

#define BLOCKS 1024
#define TPB 256
#define WPB (TPB / 32)                         // 8 waves per block
#define TOTAL_WAVES (BLOCKS * WPB)             // 8192
#define GROUPS_PER_WAVE (NGROUP / TOTAL_WAVES) // 128
#define ITERS (GROUPS_PER_WAVE / 8)            // 16 (8 groups per WMMA step)

typedef float v2f __attribute__((ext_vector_type(2)));
typedef float v8f __attribute__((ext_vector_type(8)));

// Wave-local LDS fence: same-wave LDS ops are processed in order (DScnt), so
// cross-lane RAW within one wave32 only needs the wave's own DScnt drained.
// Much cheaper than a workgroup barrier; the "memory" clobber stops the
// compiler from reordering LDS accesses across it.
#define WAVE_LDS_FENCE() asm volatile("s_wait_dscnt 0" ::: "memory")

// Composed permutation of the 22-wire CNOT ring (depth 1), flat bit k = wire 21-k.
__device__ __forceinline__ int qperm(int idx) {
  return (idx ^ (idx >> 1) ^ ((idx & 1) * 0x300000)) & 0x3FFFFF;
}

__device__ __forceinline__ float psi_lookup(const float* __restrict__ lo,
                                            const float* __restrict__ hi,
                                            int idx) {
  const int m = qperm(idx);
  return lo[m & 2047] * hi[m >> 11];
}

__global__ __launch_bounds__(TPB) void fidelity_kernel(const float* __restrict__ x,
                                                       const float* __restrict__ y,
                                                       float* __restrict__ out) {
  __shared__ float csx[2][NW];     // [0]=cos(th/2), [1]=sin(th/2)
  __shared__ float csy[2][NW];
  __shared__ float TloX[2048], ThiX[2048];   // psi_pre = Lo[m&2047]*Hi[m>>11]
  __shared__ float TloY[2048], ThiY[2048];
  __shared__ float stage[WPB][256];          // per-wave C-matrix staging

  const int tid = threadIdx.x;

  // ---- angle factors (tiny, once per block; angles ~ 0.5*N(0,1) so the fast
  // sincos path is plenty accurate) ----
  if (tid < NW) {
    float s, c;
    __sincosf(0.5f * x[tid], &s, &c);
    csx[0][tid] = c;
    csx[1][tid] = s;
  } else if (tid < 2 * NW) {
    const int w = tid - NW;
    float s, c;
    __sincosf(0.5f * y[w], &s, &c);
    csy[0][w] = c;
    csy[1][w] = s;
  }
  __syncthreads();

  // ---- build 11-bit partial-product tables ----
  for (int t = tid; t < 2048; t += TPB) {
    float lx = 1.f, hx = 1.f, ly = 1.f, hy = 1.f;
#pragma unroll
    for (int k = 0; k < 11; ++k) {
      const int b = (t >> k) & 1;
      lx *= csx[b][21 - k];   // flat bits 0..10  -> wires 21..11
      hx *= csx[b][10 - k];   // flat bits 11..21 -> wires 10..0
      ly *= csy[b][21 - k];
      hy *= csy[b][10 - k];
    }
    TloX[t] = lx; ThiX[t] = hx;
    TloY[t] = ly; ThiY[t] = hy;
  }
  __syncthreads();

  const int wid  = tid >> 5;
  const int lane = tid & 31;
  const int W    = blockIdx.x * WPB + wid;
  float* st = stage[wid];

#if __has_builtin(__builtin_amdgcn_wmma_f32_16x16x4_f32)
  // Operand-provisioning coords (lanes 0-15 carry K=0,1; lanes 16-31 are the
  // K=2,3 half of A/B and are arithmetically zeroed so EXEC stays all-ones).
  const int   l15 = lane & 15;
  const int   p   = l15 >> 1;           // group slot 0..7
  const int   tb  = l15 & 1;            // i (for A rows) / j (for B cols)
  const float msk = (lane < 16) ? 1.0f : 0.0f;
  // Extraction coords: lane writes output offset o = lane of this 32-output tile.
  const int ep = lane >> 2, ei = (lane >> 1) & 1, ej = lane & 1;
  const int em = 2 * ep + ei;           // C row
  const int en = 2 * ep + ej;           // C col
  const int saddr = (en + ((em >= 8) ? 16 : 0)) * 8 + (em & 7);

  for (int n = 0; n < ITERS; ++n) {
    const int gbase = W * GROUPS_PER_WAVE + n * 8;
    const int g = (gbase + p) << 2;     // first state index of this lane's group

    v2f A, B;
    A.x = msk * psi_lookup(TloY, ThiY, g + 2 * tb);       // A[2p+i][0] = Y_p[i,0]
    A.y = msk * psi_lookup(TloY, ThiY, g + 2 * tb + 1);   // A[2p+i][1] = Y_p[i,1]
    B.x = msk * psi_lookup(TloX, ThiX, g + tb);           // B[0][2p+j] = X_p[0,j]
    B.y = msk * psi_lookup(TloX, ThiX, g + tb + 2);       // B[1][2p+j] = X_p[1,j]

    v8f C = {0.f, 0.f, 0.f, 0.f, 0.f, 0.f, 0.f, 0.f};
    C = __builtin_amdgcn_wmma_f32_16x16x4_f32(false, A, false, B,
                                              (short)0, C, false, false);

    // Stage C and do the cross-lane diagonal-block gather. Per-wave buffer +
    // in-order same-wave LDS => wave-local DScnt fences, no workgroup barrier.
    WAVE_LDS_FENCE();                   // prior-iter gather done before overwrite
    float4* sp = (float4*)&st[lane * 8];
    sp[0] = make_float4(C[0], C[1], C[2], C[3]);
    sp[1] = make_float4(C[4], C[5], C[6], C[7]);
    WAVE_LDS_FENCE();                   // stores visible before cross-lane read

    out[(gbase << 2) + lane] = fabsf(st[saddr]);   // coalesced 128B per wave
  }
#else
  // Scalar fallback: one group of 4 outputs per thread, float4 stores.
  for (int gi = blockIdx.x * TPB + tid; gi < NGROUP; gi += BLOCKS * TPB) {
    const int g = gi << 2;
    const float y0 = psi_lookup(TloY, ThiY, g + 0);
    const float y1 = psi_lookup(TloY, ThiY, g + 1);
    const float y2 = psi_lookup(TloY, ThiY, g + 2);
    const float y3 = psi_lookup(TloY, ThiY, g + 3);
    const float x0 = psi_lookup(TloX, ThiX, g + 0);
    const float x1 = psi_lookup(TloX, ThiX, g + 1);
    const float x2 = psi_lookup(TloX, ThiX, g + 2);
    const float x3 = psi_lookup(TloX, ThiX, g + 3);
    float4 o;
    o.x = fabsf(y0 * x0 + y1 * x2);
    o.y = fabsf(y0 * x1 + y1 * x3);
    o.z = fabsf(y2 * x0 + y3 * x2);
    o.w = fabsf(y2 * x1 + y3 * x3);
    ((float4*)out)[gi] = o;
  }
#endif
}

extern "C" void kernel_launch(void* const* d_in, const int* in_sizes, int n_in,
                              void* d_out, int out_size, void* d_ws, size_t ws_size,
                              hipStream_t stream) {
  const float* x = (const float*)d_in[0];
  const float* y = (const float*)d_in[1];
  float* out = (float*)d_out;
  (void)in_sizes; (void)n_in; (void)out_size; (void)d_ws; (void)ws_size;
  fidelity_kernel<<<BLOCKS, TPB, 0, stream>>>(x, y, out);
}